// CriticNetwork_71442486001992
// MI455X (gfx1250) — compile-verified
//
#include <hip/hip_runtime.h>
#include <hip/hip_bf16.h>

typedef float v2f __attribute__((ext_vector_type(2)));
typedef float v8f __attribute__((ext_vector_type(8)));

// ---------------- sv tensor geometry (conv output, fp32) ----------------
// sv : [32][96][384][24]  C-contiguous
#define SV_C   32
#define SV_D   96
#define SV_H   384
#define SV_W   24
#define SV_SD  (SV_H*SV_W)         // 9216
#define SV_SC  ((long)SV_D*SV_SD)  // 884736
#define SV_TOT ((long)SV_C*SV_SC)  // 28311552

#define EIG_ITERS 60

// ---------------- workspace layout (float offsets) ----------------
#define OFF_SV    0L
#define OFF_P2    28311552L           // [32,96,6,24]  = 442368
#define OFF_Q1    28753920L           // [32,2,384,24] = 589824
#define OFF_T1    29343744L           // [32,2,6,24]   = 9216
#define OFF_T2    29352960L           // [32,2,6,2]    = 768
#define OFF_T1B   29353728L           // [32,96,6,2]   = 36864
#define OFF_T2B   29390592L           // [8,96,6,2]    = 9216
#define OFF_Q2    29399808L           // [32,2,384,2]  = 49152
#define OFF_Q3    29448960L           // [8,2,384,2]   = 12288
#define OFF_T2C   29461248L           // [8,2,6,24]    = 2304
#define OFF_XU    29463552L           // unfold scratch, 16384
#define OFF_G     29479936L           // Gram, up to 384*384 = 147456
#define OFF_GP    29627392L           // split-K partial Grams, 2359296
#define OFF_U0    31986688L           // 32*8
#define OFF_U1    31986944L           // 96*2
#define OFF_U2    31987136L           // 384*6
#define OFF_U3    31989440L           // 24*2
#define OFF_V6    31989488L           // 32*6
#define OFF_WV    31989680L           // 8
#define OFF_CORE  31989688L           // 192
#define GP_FLOATS 2359296L

// =====================================================================
// Conv 1x3x1 (4ch -> 32ch) + bias + ReLU. One thread per spatial site,
// produces all 32 output channels (coalesced stores per channel plane).
// =====================================================================
__global__ __launch_bounds__(256)
void conv_relu_kernel(const float* __restrict__ state,
                      const float* __restrict__ cw,
                      const float* __restrict__ cb,
                      float* __restrict__ sv)
{
  __shared__ float Wsh[32*4*3];
  __shared__ float Bsh[32];
  int tid = threadIdx.x;
  for (int i = tid; i < 384; i += 256) Wsh[i] = cw[i];
  if (tid < 32) Bsh[tid] = cb[tid];
  __syncthreads();

  long sp = (long)blockIdx.x * 256 + tid;      // 0 .. 884735
  int w = (int)(sp % SV_W);
  long t = sp / SV_W;
  int h = (int)(t % SV_H);
  int z = (int)(t / SV_H);

  float x[12];
  #pragma unroll
  for (int i = 0; i < 4; ++i)
    #pragma unroll
    for (int kt = 0; kt < 3; ++kt)
      x[i*3+kt] = state[(((long)i*96 + z)*386 + (h+kt))*24 + w];

  for (int c = 0; c < 32; ++c) {
    float acc = Bsh[c];
    const float* wc = &Wsh[c*12];
    #pragma unroll
    for (int j = 0; j < 12; ++j) acc = fmaf(x[j], wc[j], acc);
    sv[(long)c*SV_SC + sp] = fmaxf(acc, 0.f);
  }
}

// =====================================================================
// Gram G = X * X^T via V_WMMA_F32_16X16X4_F32 (one wave per 16x16 tile
// per K-slice; partials in private slices, no atomics -> deterministic).
// MODE 0: contiguous matrix with row stride ld.
// MODE 1/2/3: gather addressing of sv mode-n unfoldings.
// Hot loop: unconditional b64 pair loads (pairs never straddle gather
// chunks since chunk sizes are even and k is even), row-edge handled by
// a wave-uniform EDGE variant using v_cndmask on values (EXEC stays all
// ones for WMMA), 2 accumulators + x2 unroll, clamped speculative
// prefetch. Column guards live only in a tail that is dead for K%4==0.
// =====================================================================
template<int MODE>
__device__ __forceinline__ long gram_addr(int row, long col, int ld)
{
  if (MODE == 0) return (long)row * ld + col;
  if (MODE == 1) return (long)row * 9216 + (col / 9216) * 884736 + (col % 9216);
  if (MODE == 2) return (long)row * 24   + (col / 24) * 9216     + (col % 24);
  return (long)row + col * 24;                 // MODE 3
}

template<int MODE>
__device__ __forceinline__ v2f gram_load_pair(const float* __restrict__ X,
                                              int row, long ca, int ld)
{
  if (MODE == 3) {                 // stride-24 between columns: two b32
    const float* p = X + (long)row + ca * 24;
    v2f r; r.x = p[0]; r.y = p[24];
    return r;
  }
  return *(const v2f*)(X + gram_addr<MODE>(row, ca, ld));   // one b64
}

__device__ __forceinline__ v8f wmma_f32_4(v2f a, v2f b, v8f c)
{
#if __has_builtin(__builtin_amdgcn_wmma_f32_16x16x4_f32)
  return __builtin_amdgcn_wmma_f32_16x16x4_f32(false, a, false, b,
                                               (short)0, c, false, false);
#else
  c[0] += a.x*b.x + a.y*b.y;       // keep-alive fallback
  return c;
#endif
}

template<int MODE, bool EDGE>
__device__ __forceinline__ v8f gram_accumulate(const float* __restrict__ X,
                                               int rowA, int rowB, int va, int vb,
                                               long k0, long k1, long K, int ld,
                                               int koff)
{
  v8f acc0 = {0.f,0.f,0.f,0.f,0.f,0.f,0.f,0.f};
  v8f acc1 = {0.f,0.f,0.f,0.f,0.f,0.f,0.f,0.f};
  long kAll = K & ~3L;             // cols [k, k+3] all valid while k < kAll
  long kSafe = (k1 < kAll) ? k1 : kAll;
  long k = k0;
  for (; k + 8 <= kSafe; k += 8) {
    v2f a0 = gram_load_pair<MODE>(X, rowA, k + koff,     ld);
    v2f b0 = gram_load_pair<MODE>(X, rowB, k + koff,     ld);
    v2f a1 = gram_load_pair<MODE>(X, rowA, k + 4 + koff, ld);
    v2f b1 = gram_load_pair<MODE>(X, rowB, k + 4 + koff, ld);
    long cp = k + 512 + koff; if (cp > K - 2) cp = K - 2;
    __builtin_prefetch(X + gram_addr<MODE>(rowA, cp, ld), 0, 1);
    if (EDGE) {
      a0.x = va ? a0.x : 0.f; a0.y = va ? a0.y : 0.f;
      a1.x = va ? a1.x : 0.f; a1.y = va ? a1.y : 0.f;
      b0.x = vb ? b0.x : 0.f; b0.y = vb ? b0.y : 0.f;
      b1.x = vb ? b1.x : 0.f; b1.y = vb ? b1.y : 0.f;
    }
    acc0 = wmma_f32_4(a0, b0, acc0);
    acc1 = wmma_f32_4(a1, b1, acc1);
  }
  for (; k + 4 <= kSafe; k += 4) {
    v2f a = gram_load_pair<MODE>(X, rowA, k + koff, ld);
    v2f b = gram_load_pair<MODE>(X, rowB, k + koff, ld);
    if (EDGE) {
      a.x = va ? a.x : 0.f; a.y = va ? a.y : 0.f;
      b.x = vb ? b.x : 0.f; b.y = vb ? b.y : 0.f;
    }
    acc0 = wmma_f32_4(a, b, acc0);
  }
  // guarded tail: executes only when K % 4 != 0 (never for this model)
  for (; k < k1; k += 4) {
    long ca = k + koff, cb = ca + 1;
    int ga = ca < K, gb = cb < K;
    long cca = ga ? ca : (K - 1), ccb = gb ? cb : (K - 1);
    v2f a, b;
    a.x = X[gram_addr<MODE>(rowA, cca, ld)];
    a.y = X[gram_addr<MODE>(rowA, ccb, ld)];
    b.x = X[gram_addr<MODE>(rowB, cca, ld)];
    b.y = X[gram_addr<MODE>(rowB, ccb, ld)];
    a.x = (va && ga) ? a.x : 0.f; a.y = (va && gb) ? a.y : 0.f;
    b.x = (vb && ga) ? b.x : 0.f; b.y = (vb && gb) ? b.y : 0.f;
    acc0 = wmma_f32_4(a, b, acc0);
  }
  acc0 += acc1;
  return acc0;
}

template<int MODE>
__global__ __launch_bounds__(32)
void gram_wmma_kernel(const float* __restrict__ X, float* __restrict__ Gp,
                      int d, long K, int ld, long kPerSplit)
{
  // decode upper-triangular tile index
  int ntile = (d + 15) >> 4;
  int lin = blockIdx.x;
  int it = 0, rowlen = ntile;
  while (lin >= rowlen) { lin -= rowlen; ++it; --rowlen; }
  int jt = it + lin;

  int lane = threadIdx.x;
  int m    = lane & 15;            // M row within tile
  int koff = (lane >> 4) << 1;     // lanes 0-15: K 0,1 ; lanes 16-31: K 2,3
  int rowA = it * 16 + m;
  int rowB = jt * 16 + m;
  int va = rowA < d;
  int vb = rowB < d;
  int rowAc = va ? rowA : d - 1;   // clamped: loads always in-bounds
  int rowBc = vb ? rowB : d - 1;

  long k0 = (long)blockIdx.y * kPerSplit;
  long k1 = k0 + kPerSplit; if (k1 > K) k1 = K;
  if (k1 < k0) k1 = k0;

  bool edge = ((it * 16 + 15) >= d) || ((jt * 16 + 15) >= d);  // wave-uniform
  v8f acc;
  if (edge) acc = gram_accumulate<MODE, true >(X, rowAc, rowBc, va, vb, k0, k1, K, ld, koff);
  else      acc = gram_accumulate<MODE, false>(X, rowAc, rowBc, 1, 1,  k0, k1, K, ld, koff);

  float* Gs = Gp + (long)blockIdx.y * d * d;
  int nlo = lane & 15;
  int mhi = (lane >> 4) * 8;       // C layout: VGPR r -> M = r (+8 for hi lanes)
  #pragma unroll
  for (int rr = 0; rr < 8; ++rr) {
    int row = it * 16 + rr + mhi;
    int col = jt * 16 + nlo;
    if (row < d && col < d) {
      Gs[row * d + col] = acc[rr];
      if (it != jt) Gs[col * d + row] = acc[rr];   // symmetric mirror
    }
  }
}

__global__ void gram_reduce_kernel(const float* __restrict__ Gp,
                                   float* __restrict__ G, int dd, int ks)
{
  int e = blockIdx.x * blockDim.x + threadIdx.x;
  if (e < dd) {
    float s = 0.f;
    for (int k = 0; k < ks; ++k) s += Gp[(long)k * dd + e];
    G[e] = s;
  }
}

// =====================================================================
// Generic n-mode product: out = in x_mode U^T  (U is K x R, row major),
// result axis placed back at `mode` (matches _mode_dot_t).
// =====================================================================
__global__ void mode_dot_kernel(const float* __restrict__ in,
                                const float* __restrict__ U,
                                float* __restrict__ out,
                                int D0, int D1, int D2, int D3,
                                int mode, int R, long outTotal)
{
  long idx = (long)blockIdx.x * blockDim.x + threadIdx.x;
  if (idx >= outTotal) return;
  int E0=D0,E1=D1,E2=D2,E3=D3;
  if (mode==0) E0=R; else if (mode==1) E1=R; else if (mode==2) E2=R; else E3=R;
  long t = idx;
  int o3 = (int)(t % E3); t /= E3;
  int o2 = (int)(t % E2); t /= E2;
  int o1 = (int)(t % E1); t /= E1;
  int o0 = (int)t;
  int r  = (mode==0)?o0:(mode==1)?o1:(mode==2)?o2:o3;
  long s3 = 1, s2 = D3, s1 = (long)D2*D3, s0 = (long)D1*D2*D3;
  long stride; int K;
  int i0=o0,i1=o1,i2=o2,i3=o3;
  if      (mode==0){ i0=0; stride=s0; K=D0; }
  else if (mode==1){ i1=0; stride=s1; K=D1; }
  else if (mode==2){ i2=0; stride=s2; K=D2; }
  else             { i3=0; stride=s3; K=D3; }
  const float* p = in + (long)i0*s0 + (long)i1*s1 + (long)i2*s2 + (long)i3*s3;
  float acc = 0.f;
  for (int k = 0; k < K; ++k) acc = fmaf(p[(long)k*stride], U[(long)k*R + r], acc);
  out[idx] = acc;
}

// =====================================================================
// Materialize unfolding of a small 4-D tensor. trans=1 writes X^T.
// =====================================================================
__global__ void unfold_kernel(const float* __restrict__ in, float* __restrict__ out,
                              int D0,int D1,int D2,int D3,int mode,int trans,long total)
{
  long idx = (long)blockIdx.x * blockDim.x + threadIdx.x;
  if (idx >= total) return;
  int dims[4] = {D0,D1,D2,D3};
  long t = idx;
  int i3=(int)(t%D3); t/=D3; int i2=(int)(t%D2); t/=D2; int i1=(int)(t%D1); t/=D1;
  int ii[4] = {(int)t,i1,i2,i3};
  int j = ii[mode];
  long k = 0;
  for (int m = 0; m < 4; ++m) if (m != mode) k = k*dims[m] + ii[m];
  long d = dims[mode], K = total / d;
  out[trans ? (k*d + j) : ((long)j*K + k)] = in[idx];
}

// =====================================================================
// Top-r eigenpairs of symmetric PSD G (n<=384, r<=8): subspace iteration
// with modified Gram-Schmidt, then Rayleigh-Ritz (in-thread rxr Jacobi),
// eigenvalues descending. One workgroup; G stays L2-resident.
// =====================================================================
__global__ __launch_bounds__(256)
void eig_topr_kernel(const float* __restrict__ G, int n, int r,
                     float* __restrict__ U, float* __restrict__ wout, int iters)
{
  __shared__ float Q[384*8];
  __shared__ float Y[384*8];
  __shared__ float red[256];
  __shared__ float Bm[64];
  __shared__ float Wm[64];
  __shared__ float ev[8];
  int tid = threadIdx.x;

  for (long idx = tid; idx < (long)n*r; idx += 256) {
    int i = (int)(idx / r), j = (int)(idx % r);
    unsigned hsh = (unsigned)i*2654435761u + (unsigned)j*40503u + 12345u;
    Q[idx] = ((i % n) == j ? 1.f : 0.f) + (float)((hsh >> 16) & 255) * 4e-7f;
  }
  __syncthreads();

  for (int it = 0; it <= iters; ++it) {
    // Y = G * Q
    for (int i = tid; i < n; i += 256) {
      float acc[8];
      #pragma unroll
      for (int j = 0; j < 8; ++j) acc[j] = 0.f;
      const float* Gi = G + (long)i*n;
      for (int k = 0; k < n; ++k) {
        float g = Gi[k];
        const float* Qk = &Q[k*r];
        for (int j = 0; j < r; ++j) acc[j] = fmaf(g, Qk[j], acc[j]);
      }
      for (int j = 0; j < r; ++j) Y[i*r + j] = acc[j];
    }
    __syncthreads();
    if (it == iters) break;          // keep Y = G*Q of orthonormal Q

    // modified Gram-Schmidt on the r columns of Y
    for (int j = 0; j < r; ++j) {
      for (int p = 0; p < j; ++p) {
        float part = 0.f;
        for (int i = tid; i < n; i += 256) part += Y[i*r+p]*Y[i*r+j];
        red[tid] = part; __syncthreads();
        for (int s = 128; s > 0; s >>= 1) {
          if (tid < s) red[tid] += red[tid+s];
          __syncthreads();
        }
        float dot = red[0]; __syncthreads();
        for (int i = tid; i < n; i += 256) Y[i*r+j] -= dot * Y[i*r+p];
        __syncthreads();
      }
      float part = 0.f;
      for (int i = tid; i < n; i += 256) part += Y[i*r+j]*Y[i*r+j];
      red[tid] = part; __syncthreads();
      for (int s = 128; s > 0; s >>= 1) {
        if (tid < s) red[tid] += red[tid+s];
        __syncthreads();
      }
      float inv = rsqrtf(fmaxf(red[0], 1e-30f)); __syncthreads();
      for (int i = tid; i < n; i += 256) Y[i*r+j] *= inv;
      __syncthreads();
    }
    for (long idx = tid; idx < (long)n*r; idx += 256) Q[idx] = Y[idx];
    __syncthreads();
  }

  // B = Q^T * Y  (r x r)
  if (tid < r*r) {
    int p = tid / r, q = tid % r;
    float s = 0.f;
    for (int i = 0; i < n; ++i) s += Q[i*r+p]*Y[i*r+q];
    Bm[p*r+q] = s;
  }
  __syncthreads();

  if (tid == 0) {
    float A[8][8], V[8][8];
    for (int p = 0; p < r; ++p)
      for (int q = 0; q < r; ++q) { A[p][q] = Bm[p*r+q]; V[p][q] = (p==q)?1.f:0.f; }
    for (int sw = 0; sw < 30; ++sw) {
      for (int p = 0; p < r-1; ++p)
        for (int q = p+1; q < r; ++q) {
          float apq = A[p][q];
          if (fabsf(apq) < 1e-12f) continue;
          float tau = (A[q][q] - A[p][p]) / (2.f*apq);
          float tt  = (tau >= 0.f ? 1.f : -1.f) / (fabsf(tau) + sqrtf(1.f + tau*tau));
          float c   = rsqrtf(1.f + tt*tt), s2 = tt*c;
          for (int k = 0; k < r; ++k) { float x=A[k][p], y=A[k][q];
            A[k][p]=c*x - s2*y; A[k][q]=s2*x + c*y; }
          for (int k = 0; k < r; ++k) { float x=A[p][k], y=A[q][k];
            A[p][k]=c*x - s2*y; A[q][k]=s2*x + c*y; }
          for (int k = 0; k < r; ++k) { float x=V[k][p], y=V[k][q];
            V[k][p]=c*x - s2*y; V[k][q]=s2*x + c*y; }
        }
    }
    int ord[8];
    for (int j = 0; j < r; ++j) ord[j] = j;
    for (int a = 0; a < r; ++a) {
      int best = a;
      for (int b = a+1; b < r; ++b)
        if (A[ord[b]][ord[b]] > A[ord[best]][ord[best]]) best = b;
      int tmp = ord[a]; ord[a] = ord[best]; ord[best] = tmp;
    }
    for (int j = 0; j < r; ++j) {
      ev[j] = A[ord[j]][ord[j]];
      for (int p = 0; p < r; ++p) Wm[p*r+j] = V[p][ord[j]];
    }
  }
  __syncthreads();

  // U = Q * W
  for (long idx = tid; idx < (long)n*r; idx += 256) {
    int i = (int)(idx / r), j = (int)(idx % r);
    float s = 0.f;
    for (int p = 0; p < r; ++p) s = fmaf(Q[i*r+p], Wm[p*r+j], s);
    U[idx] = s;
  }
  if (tid < r) wout[tid] = ev[tid];
}

// U2[h][r] = sum_i Xt[i][h] * v[i][r] / sqrt(max(w,1e-12))  (d>k branch)
__global__ void mode2_factor_kernel(const float* __restrict__ Xt,  // 32 x 384
                                    const float* __restrict__ v,   // 32 x 6
                                    const float* __restrict__ w,   // 6
                                    float* __restrict__ U2)        // 384 x 6
{
  int h = blockIdx.x * blockDim.x + threadIdx.x;
  if (h >= 384) return;
  for (int rr = 0; rr < 6; ++rr) {
    float s = 0.f;
    for (int i = 0; i < 32; ++i) s = fmaf(Xt[(long)i*384 + h], v[i*6 + rr], s);
    U2[h*6 + rr] = s * rsqrtf(fmaxf(w[rr], 1e-12f));
  }
}

// head: out = relu(core + relu(wa@action+ba)) . wq + bq
__global__ __launch_bounds__(192)
void final_kernel(const float* __restrict__ core, const float* __restrict__ action,
                  const float* __restrict__ wa, const float* __restrict__ ba,
                  const float* __restrict__ wq, const float* __restrict__ bq,
                  float* __restrict__ out)
{
  __shared__ float red[192];
  int i = threadIdx.x;
  float acc = ba[i];
  for (int j = 0; j < 96; ++j) acc = fmaf(wa[i*96 + j], action[j], acc);
  float sav = fmaxf(core[i] + fmaxf(acc, 0.f), 0.f);
  red[i] = sav * wq[i];
  __syncthreads();
  if (i < 64) red[i] += red[i+64] + red[i+128];
  __syncthreads();
  for (int s = 32; s > 0; s >>= 1) {
    if (i < s) red[i] += red[i+s];
    __syncthreads();
  }
  if (i == 0) out[0] = red[0] + bq[0];
}

// ---------------------------------------------------------------------
// host-side helpers
// ---------------------------------------------------------------------
static inline int pick_ksplit(int d, long K) {
  long dd = (long)d*d;
  long maxs = GP_FLOATS / dd; if (maxs < 1) maxs = 1;
  long ks = K / 2048; if (ks < 1) ks = 1;
  if (ks > maxs) ks = maxs;
  if (ks > 256)  ks = 256;
  return (int)ks;
}

template<int MODE>
static void launch_gram(const float* X, float* G, float* Gp,
                        int d, long K, int ld, hipStream_t s) {
  int ks = pick_ksplit(d, K);
  long kps = (((K + ks - 1) / ks) + 3) & ~3L;
  int ntile = (d + 15) / 16;
  int tris  = ntile * (ntile + 1) / 2;
  gram_wmma_kernel<MODE><<<dim3((unsigned)tris, (unsigned)ks), 32, 0, s>>>(X, Gp, d, K, ld, kps);
  int dd = d * d;
  gram_reduce_kernel<<<(unsigned)((dd + 255) / 256), 256, 0, s>>>(Gp, G, dd, ks);
}

static void launch_mode_dot(const float* in, const float* U, float* out,
                            int D0,int D1,int D2,int D3,int mode,int R, hipStream_t s) {
  int Dm = (mode==0)?D0:(mode==1)?D1:(mode==2)?D2:D3;
  long total = (long)D0*D1*D2*D3 / Dm * R;
  mode_dot_kernel<<<(unsigned)((total + 255) / 256), 256, 0, s>>>(in, U, out,
                                                                  D0,D1,D2,D3, mode, R, total);
}

static void launch_unfold(const float* in, float* out,
                          int D0,int D1,int D2,int D3,int mode,int trans, hipStream_t s) {
  long total = (long)D0*D1*D2*D3;
  unfold_kernel<<<(unsigned)((total + 255) / 256), 256, 0, s>>>(in, out,
                                                                D0,D1,D2,D3, mode, trans, total);
}

extern "C" void kernel_launch(void* const* d_in, const int* in_sizes, int n_in,
                              void* d_out, int out_size, void* d_ws, size_t ws_size,
                              hipStream_t stream)
{
  (void)in_sizes; (void)n_in; (void)out_size; (void)ws_size;
  const float* state  = (const float*)d_in[0];
  const float* action = (const float*)d_in[1];
  const float* conv_w = (const float*)d_in[2];
  const float* conv_b = (const float*)d_in[3];
  const float* wa     = (const float*)d_in[4];
  const float* ba     = (const float*)d_in[5];
  const float* wq     = (const float*)d_in[6];
  const float* bq     = (const float*)d_in[7];

  float* F   = (float*)d_ws;
  float* sv  = F + OFF_SV;
  float* P2  = F + OFF_P2;
  float* Q1  = F + OFF_Q1;
  float* T1  = F + OFF_T1;
  float* T2  = F + OFF_T2;
  float* T1B = F + OFF_T1B;
  float* T2B = F + OFF_T2B;
  float* Q2  = F + OFF_Q2;
  float* Q3  = F + OFF_Q3;
  float* T2C = F + OFF_T2C;
  float* XU  = F + OFF_XU;
  float* G   = F + OFF_G;
  float* GP  = F + OFF_GP;
  float* U0  = F + OFF_U0;
  float* U1  = F + OFF_U1;
  float* U2  = F + OFF_U2;
  float* U3  = F + OFF_U3;
  float* V6  = F + OFF_V6;
  float* WV  = F + OFF_WV;
  float* CR  = F + OFF_CORE;

  // 1) conv + bias + relu  -> sv [32,96,384,24]
  conv_relu_kernel<<<3456, 256, 0, stream>>>(state, conv_w, conv_b, sv);

  // 2) HOSVD init: mode Grams (WMMA f32) + top-r eigvecs
  launch_gram<0>(sv, G, GP, 32, 884736L, 884736, stream);
  eig_topr_kernel<<<1, 256, 0, stream>>>(G, 32, 8, U0, WV, EIG_ITERS);
  launch_gram<1>(sv, G, GP, 96, 294912L, 0, stream);
  eig_topr_kernel<<<1, 256, 0, stream>>>(G, 96, 2, U1, WV, EIG_ITERS);
  launch_gram<2>(sv, G, GP, 384, 73728L, 0, stream);
  eig_topr_kernel<<<1, 256, 0, stream>>>(G, 384, 6, U2, WV, EIG_ITERS);
  launch_gram<3>(sv, G, GP, 24, 1179648L, 0, stream);
  eig_topr_kernel<<<1, 256, 0, stream>>>(G, 24, 2, U3, WV, EIG_ITERS);

  // 3) HOOI sweeps. P2 = sv x2 U2^T is reused across skip0/skip1/next sweep.
  for (int sweep = 0; sweep < 5; ++sweep) {
    if (sweep == 0)
      launch_mode_dot(sv, U2, P2, 32,96,384,24, 2, 6, stream);   // [32,96,6,24]

    // skip = 0 : (x2 U2)(x1 U1)(x3 U3); unfold0 contiguous -> G 32x32, top 8
    launch_mode_dot(P2, U1, T1, 32,96,6,24, 1, 2, stream);       // [32,2,6,24]
    launch_mode_dot(T1, U3, T2, 32,2,6,24, 3, 2, stream);        // [32,2,6,2]
    launch_gram<0>(T2, G, GP, 32, 24L, 24, stream);
    eig_topr_kernel<<<1, 256, 0, stream>>>(G, 32, 8, U0, WV, EIG_ITERS);

    // skip = 1 : (x2 U2)(x3 U3)(x0 U0); unfold1 -> G 96x96, top 2
    launch_mode_dot(P2, U3, T1B, 32,96,6,24, 3, 2, stream);      // [32,96,6,2]
    launch_mode_dot(T1B, U0, T2B, 32,96,6,2, 0, 8, stream);      // [8,96,6,2]
    launch_unfold(T2B, XU, 8,96,6,2, 1, 0, stream);              // 96 x 96
    launch_gram<0>(XU, G, GP, 96, 96L, 96, stream);
    eig_topr_kernel<<<1, 256, 0, stream>>>(G, 96, 2, U1, WV, EIG_ITERS);

    // skip = 2 : (x1 U1)(x3 U3)(x0 U0); unfold2 is 384x32 (d>k): G=X^T X
    launch_mode_dot(sv, U1, Q1, 32,96,384,24, 1, 2, stream);     // [32,2,384,24]
    launch_mode_dot(Q1, U3, Q2, 32,2,384,24, 3, 2, stream);      // [32,2,384,2]
    launch_mode_dot(Q2, U0, Q3, 32,2,384,2, 0, 8, stream);       // [8,2,384,2]
    launch_unfold(Q3, XU, 8,2,384,2, 2, 1, stream);              // X^T : 32 x 384
    launch_gram<0>(XU, G, GP, 32, 384L, 384, stream);            // G = X^T X 32x32
    eig_topr_kernel<<<1, 256, 0, stream>>>(G, 32, 6, V6, WV, EIG_ITERS);
    mode2_factor_kernel<<<2, 256, 0, stream>>>(XU, V6, WV, U2);  // U2 = X v / sqrt(w)

    // skip = 3 : (x2 U2new)(x1 U1)(x0 U0); unfold3 -> G 24x24, top 2
    launch_mode_dot(sv, U2, P2, 32,96,384,24, 2, 6, stream);     // refresh P2
    launch_mode_dot(P2, U1, T1, 32,96,6,24, 1, 2, stream);       // [32,2,6,24]
    launch_mode_dot(T1, U0, T2C, 32,2,6,24, 0, 8, stream);       // [8,2,6,24]
    launch_unfold(T2C, XU, 8,2,6,24, 3, 0, stream);              // 24 x 96
    launch_gram<0>(XU, G, GP, 24, 96L, 96, stream);
    eig_topr_kernel<<<1, 256, 0, stream>>>(G, 24, 2, U3, WV, EIG_ITERS);
  }

  // 4) core = P2 (final U2) x1 U1 x3 U3 x0 U0  -> [8,2,6,2] flat 192
  launch_mode_dot(P2, U1, T1, 32,96,6,24, 1, 2, stream);
  launch_mode_dot(T1, U3, T2, 32,2,6,24, 3, 2, stream);
  launch_mode_dot(T2, U0, CR, 32,2,6,2, 0, 8, stream);

  // 5) head
  final_kernel<<<1, 192, 0, stream>>>(CR, action, wa, ba, wq, bq, (float*)d_out);
}